// Model_79937931313419
// MI455X (gfx1250) — compile-verified
//
#include <hip/hip_runtime.h>
#include <hip/hip_bf16.h>
#include <stdint.h>

// ---------------------------------------------------------------------------
// CDNA5 (gfx1250) implementation of the hetero-GraphSAGE encoder/decoder.
// Dense math runs through V_WMMA_F32_16X16X4_F32 (fp32 in/out, matches the
// fp32 reference). Scatter (segment-mean) uses hardware fp32 global atomics.
// ---------------------------------------------------------------------------

typedef float v2f __attribute__((ext_vector_type(2)));
typedef float v8f __attribute__((ext_vector_type(8)));

#define HID 128

__device__ __forceinline__ void atom_add_f32(float* p, float v) {
    // unsafeAtomicAdd lowers to global_atomic_add_f32 on AMD (no CAS loop).
    unsafeAtomicAdd(p, v);
}

__device__ __forceinline__ v8f wmma_f32(v2f a, v2f b, v8f c) {
    return __builtin_amdgcn_wmma_f32_16x16x4_f32(false, a, false, b,
                                                 (short)0, c, false, false);
}

// ---------------------------------------------------------------------------
// Degree count: deg[dst[e]] += 1
// ---------------------------------------------------------------------------
__global__ void degree_kernel(const int* __restrict__ dst, float* __restrict__ deg,
                              int nedges) {
    int e = blockIdx.x * blockDim.x + threadIdx.x;
    if (e < nedges) atom_add_f32(&deg[dst[e]], 1.0f);
}

// ---------------------------------------------------------------------------
// Scatter-add of 128-float feature rows:
//   agg[dst[e], :] += xsrc[xids ? xids[src[e]] : src[e], :]
// 32 lanes per edge, 4 floats (16B) per lane -> 512B coalesced row reads.
// ---------------------------------------------------------------------------
__global__ void scatter_add_kernel(const float* __restrict__ xsrc,
                                   const int* __restrict__ xids,
                                   const int* __restrict__ src,
                                   const int* __restrict__ dst,
                                   float* __restrict__ agg, int nedges) {
    long long tid = (long long)blockIdx.x * blockDim.x + threadIdx.x;
    int e = (int)(tid >> 5);
    if (e >= nedges) return;
    int j = ((int)tid & 31) << 2;           // float offset within the row
    int s = src[e];
    if (xids) s = xids[s];
    int d = dst[e];
    const float4 v = *(const float4*)(xsrc + (size_t)s * HID + j);
    float* p = agg + (size_t)d * HID + j;
    atom_add_f32(p + 0, v.x);
    atom_add_f32(p + 1, v.y);
    atom_add_f32(p + 2, v.z);
    atom_add_f32(p + 3, v.w);
}

// ---------------------------------------------------------------------------
// Fused SAGE layer:
//   out[i,n] = act( (agg[i,:]/max(deg[i],1)) @ WL  +  x[gid(i),:] @ WR + bias )
//
// Block: 256 threads (8 wave32), 32 output rows per block.
// NC==128: wave w -> column tile w, two 16-row sub-tiles (B frags reused 2x).
// NC==64 : waves split 2 row-groups x 4 column tiles, one sub-tile each.
// NC is compile-time so all B loads are base + immediate offset (no 64-bit
// VALU address math in the WMMA loop).
//
// V_WMMA_F32_16X16X4_F32 fragment layout (per ISA):
//   A 16x4: lane L<16 -> {A[M=L][k],A[M=L][k+1]}, L>=16 -> {k+2,k+3}
//   B 4x16: lane L<16 -> {B[k][N=L],B[k+1][N=L]}, L>=16 -> {k+2,k+3}
//   C 16x16: vgpr r, lane L -> row r + 8*(L>=16), col L&15
// ---------------------------------------------------------------------------
template <int NC, bool HAS_WL>
__global__ void sage_gemm_kernel(const float* __restrict__ agg,
                                 const float* __restrict__ deg,
                                 const float* __restrict__ x,
                                 const int* __restrict__ xids,
                                 const float* __restrict__ WL,
                                 const float* __restrict__ WR,
                                 const float* __restrict__ bias,
                                 float* __restrict__ out,
                                 int N, int do_relu) {
    constexpr int NTILES = NC / 16;   // column tiles (8 or 4)
    constexpr int RG     = 8 / NTILES; // wave row-groups (1 or 2)
    constexpr int MT     = 2 / RG;     // 16-row sub-tiles per wave (2 or 1)

    __shared__ float sA[32][HID + 2];  // scaled aggregate rows (+2: no bank conflict)
    __shared__ float sX[32][HID + 2];  // destination-feature rows

    const int row0 = blockIdx.x << 5;
    const int tid  = threadIdx.x;

    // ---- stage 32 rows of (agg/deg) and x into LDS (coalesced) ----
    for (int idx = tid; idx < 32 * HID; idx += 256) {
        int r = idx >> 7, c = idx & (HID - 1);
        int row = row0 + r;
        float av = 0.f, xv = 0.f;
        if (row < N) {
            if (HAS_WL) av = agg[(size_t)row * HID + c] / fmaxf(deg[row], 1.0f);
            int xr = xids ? xids[row] : row;
            xv = x[(size_t)xr * HID + c];
        }
        sA[r][c] = av;
        sX[r][c] = xv;
    }
    __syncthreads();

    const int wave = tid >> 5;
    const int colt = wave % NTILES;
    const int rowg = wave / NTILES;
    const int lane = tid & 31;
    const int l15  = lane & 15;
    const int half = lane >> 4;          // 0: K={k,k+1}, 1: K={k+2,k+3}
    const int n    = (colt << 4) + l15;  // this lane's output column
    const int koff = 2 * half;
    const int m0   = rowg * (16 * MT);   // first LDS row owned by this wave

    v8f acc[MT];
#pragma unroll
    for (int t = 0; t < MT; ++t) acc[t] = {};

    if (HAS_WL) {
        const float* WLp = WL + (size_t)koff * NC + n;  // per-lane base
#pragma unroll
        for (int k = 0; k < HID; k += 4) {
            v2f b = { WLp[(size_t)k * NC], WLp[(size_t)(k + 1) * NC] };
#pragma unroll
            for (int t = 0; t < MT; ++t) {
                const int ar = m0 + 16 * t + l15;
                v2f a = { sA[ar][k + koff], sA[ar][k + koff + 1] };
                acc[t] = wmma_f32(a, b, acc[t]);
            }
        }
    }
    {
        const float* WRp = WR + (size_t)koff * NC + n;
#pragma unroll
        for (int k = 0; k < HID; k += 4) {
            v2f b = { WRp[(size_t)k * NC], WRp[(size_t)(k + 1) * NC] };
#pragma unroll
            for (int t = 0; t < MT; ++t) {
                const int ar = m0 + 16 * t + l15;
                v2f a = { sX[ar][k + koff], sX[ar][k + koff + 1] };
                acc[t] = wmma_f32(a, b, acc[t]);
            }
        }
    }

    const float bn = bias[n];
#pragma unroll
    for (int t = 0; t < MT; ++t) {
#pragma unroll
        for (int r = 0; r < 8; ++r) {
            int row = row0 + m0 + 16 * t + r + 8 * half;
            float v = acc[t][r] + bn;
            if (do_relu) v = fmaxf(v, 0.0f);
            if (row < N) out[(size_t)row * NC + n] = v;
        }
    }
}

// ---------------------------------------------------------------------------
// Edge decoder: per edge e
//   z = [z_sotu[row[e]] | z_taxon[col[e]]]      (128)
//   h = relu(z @ W1 + b1)                        (64)
//   out[e] = h . w2 + b2
// Block: 256 threads (8 wave32), 32 edges per block.
// Waves: 2 row-groups x 4 column tiles; W1 GEMM via WMMA.
// ---------------------------------------------------------------------------
__global__ void decoder_kernel(const float* __restrict__ zs,
                               const float* __restrict__ zt,
                               const int* __restrict__ erow,
                               const int* __restrict__ ecol,
                               const float* __restrict__ W1,
                               const float* __restrict__ b1,
                               const float* __restrict__ w2,
                               const float* __restrict__ b2,
                               float* __restrict__ out, int nedges) {
    __shared__ float sA[32][HID + 2];
    __shared__ float sH[32][64 + 2];

    const int e0  = blockIdx.x << 5;
    const int tid = threadIdx.x;

    for (int idx = tid; idx < 32 * HID; idx += 256) {
        int r = idx >> 7, c = idx & (HID - 1);
        int e = e0 + r;
        float v = 0.f;
        if (e < nedges)
            v = (c < 64) ? zs[(size_t)erow[e] * 64 + c]
                         : zt[(size_t)ecol[e] * 64 + (c - 64)];
        sA[r][c] = v;
    }
    __syncthreads();

    const int wave = tid >> 5;
    const int colt = wave & 3;
    const int rowg = wave >> 2;
    const int lane = tid & 31;
    const int l15  = lane & 15;
    const int half = lane >> 4;
    const int n    = (colt << 4) + l15;
    const int koff = 2 * half;
    const int m0   = rowg << 4;

    v8f acc = {};
    const float* W1p = W1 + (size_t)koff * 64 + n;
#pragma unroll
    for (int k = 0; k < HID; k += 4) {
        v2f b = { W1p[(size_t)k * 64], W1p[(size_t)(k + 1) * 64] };
        v2f a = { sA[m0 + l15][k + koff], sA[m0 + l15][k + koff + 1] };
        acc = wmma_f32(a, b, acc);
    }
    const float bn = b1[n];
#pragma unroll
    for (int r = 0; r < 8; ++r)
        sH[m0 + r + 8 * half][n] = fmaxf(acc[r] + bn, 0.0f);
    __syncthreads();

    if (tid < 32) {
        int e = e0 + tid;
        if (e < nedges) {
            float s = b2[0];
#pragma unroll 8
            for (int j = 0; j < 64; ++j) s += sH[tid][j] * w2[j];
            out[e] = s;
        }
    }
}

// ---------------------------------------------------------------------------
// Host orchestration
// ---------------------------------------------------------------------------
static inline void run_sage128(const float* agg, const float* deg, const float* x,
                               const int* xids, const float* WL, const float* WR,
                               const float* bias, float* out, int N, bool relu,
                               hipStream_t stream) {
    dim3 grid((N + 31) / 32), block(256);
    sage_gemm_kernel<128, true><<<grid, block, 0, stream>>>(
        agg, deg, x, xids, WL, WR, bias, out, N, relu ? 1 : 0);
}

static inline void run_proj64(const float* x, const float* W, const float* bias,
                              float* out, int N, hipStream_t stream) {
    dim3 grid((N + 31) / 32), block(256);
    sage_gemm_kernel<64, false><<<grid, block, 0, stream>>>(
        nullptr, nullptr, x, nullptr, nullptr, W, bias, out, N, 0);
}

static inline void run_scatter(const float* xsrc, const int* xids, const int* src,
                               const int* dst, float* agg, int nedges,
                               hipStream_t stream) {
    long long threads = (long long)nedges * 32;
    dim3 grid((unsigned)((threads + 255) / 256)), block(256);
    scatter_add_kernel<<<grid, block, 0, stream>>>(xsrc, xids, src, dst, agg, nedges);
}

extern "C" void kernel_launch(void* const* d_in, const int* in_sizes, int n_in,
                              void* d_out, int out_size, void* d_ws, size_t ws_size,
                              hipStream_t stream) {
    (void)n_in; (void)out_size; (void)ws_size;

    const int*   sotu_ids  = (const int*)d_in[0];
    const int*   taxon_ids = (const int*)d_in[1];
    const int*   ett_src   = (const int*)d_in[2];
    const int*   ett_dst   = (const int*)d_in[3];
    const int*   ets_src   = (const int*)d_in[4];
    const int*   ets_dst   = (const int*)d_in[5];
    const int*   elbl_row  = (const int*)d_in[6];
    const int*   elbl_col  = (const int*)d_in[7];
    const float* sotu_emb  = (const float*)d_in[8];
    const float* taxon_emb = (const float*)d_in[9];
    const float *s1_wl = (const float*)d_in[10], *s1_bl = (const float*)d_in[11], *s1_wr = (const float*)d_in[12];
    const float *s2_wl = (const float*)d_in[13], *s2_bl = (const float*)d_in[14], *s2_wr = (const float*)d_in[15];
    const float *s3_wl = (const float*)d_in[16], *s3_bl = (const float*)d_in[17], *s3_wr = (const float*)d_in[18];
    const float *sl_w  = (const float*)d_in[19], *sl_b  = (const float*)d_in[20];
    const float *t1_wl = (const float*)d_in[21], *t1_bl = (const float*)d_in[22], *t1_wr = (const float*)d_in[23];
    const float *t2_wl = (const float*)d_in[24], *t2_bl = (const float*)d_in[25], *t2_wr = (const float*)d_in[26];
    const float *tl_w  = (const float*)d_in[27], *tl_b  = (const float*)d_in[28];
    const float *d1_w  = (const float*)d_in[29], *d1_b  = (const float*)d_in[30];
    const float *d2_w  = (const float*)d_in[31], *d2_b  = (const float*)d_in[32];

    const int N_SOTU  = in_sizes[0];
    const int N_TAXON = in_sizes[1];
    const int E_TT    = in_sizes[2];
    const int E_TS    = in_sizes[4];
    const int E_LBL   = in_sizes[6];

    // ---- workspace carve-out ----
    size_t off = 0;
    auto carve = [&](size_t bytes) -> void* {
        void* p = (char*)d_ws + off;
        off += (bytes + 255) & ~(size_t)255;
        return p;
    };
    float* agg_tt  = (float*)carve((size_t)N_TAXON * HID * 4); // shared by s1/t1, reused for t2
    float* agg_ts  = (float*)carve((size_t)N_SOTU  * HID * 4); // s2, reused for s3
    float* taxon_x = (float*)carve((size_t)N_TAXON * HID * 4);
    float* tx      = (float*)carve((size_t)N_TAXON * HID * 4); // taxon-encoder hidden (in-place t2)
    float* sotu_x  = (float*)carve((size_t)N_SOTU  * HID * 4); // sotu-encoder hidden (in-place s3)
    float* z_sotu  = (float*)carve((size_t)N_SOTU  * 64 * 4);
    float* z_taxon = (float*)carve((size_t)N_TAXON * 64 * 4);
    float* deg_tt  = (float*)carve((size_t)N_TAXON * 4);
    float* deg_ts  = (float*)carve((size_t)N_SOTU  * 4);

    // ---- degrees (computed once; ett deg shared by s1/t1/t2, ets by s2/s3) ----
    hipMemsetAsync(deg_tt, 0, (size_t)N_TAXON * 4, stream);
    hipMemsetAsync(deg_ts, 0, (size_t)N_SOTU * 4, stream);
    degree_kernel<<<(E_TT + 255) / 256, 256, 0, stream>>>(ett_dst, deg_tt, E_TT);
    degree_kernel<<<(E_TS + 255) / 256, 256, 0, stream>>>(ets_dst, deg_ts, E_TS);

    // ---- layer-1 aggregates ----
    hipMemsetAsync(agg_tt, 0, (size_t)N_TAXON * HID * 4, stream);
    hipMemsetAsync(agg_ts, 0, (size_t)N_SOTU * HID * 4, stream);
    run_scatter(taxon_emb, taxon_ids, ett_src, ett_dst, agg_tt, E_TT, stream); // s1 & t1
    run_scatter(taxon_emb, taxon_ids, ets_src, ets_dst, agg_ts, E_TS, stream); // s2

    // ---- layer-1 SAGE GEMMs ----
    run_sage128(agg_tt, deg_tt, taxon_emb, taxon_ids, s1_wl, s1_wr, s1_bl,
                taxon_x, N_TAXON, true, stream);
    run_sage128(agg_tt, deg_tt, taxon_emb, taxon_ids, t1_wl, t1_wr, t1_bl,
                tx, N_TAXON, true, stream);
    run_sage128(agg_ts, deg_ts, sotu_emb, sotu_ids, s2_wl, s2_wr, s2_bl,
                sotu_x, N_SOTU, true, stream);

    // ---- layer-2: sotu branch (agg over taxon_x, in-place update of sotu_x) ----
    hipMemsetAsync(agg_ts, 0, (size_t)N_SOTU * HID * 4, stream);
    run_scatter(taxon_x, nullptr, ets_src, ets_dst, agg_ts, E_TS, stream);
    run_sage128(agg_ts, deg_ts, sotu_x, nullptr, s3_wl, s3_wr, s3_bl,
                sotu_x, N_SOTU, true, stream);

    // ---- layer-2: taxon branch (agg over tx, in-place update of tx) ----
    hipMemsetAsync(agg_tt, 0, (size_t)N_TAXON * HID * 4, stream);
    run_scatter(tx, nullptr, ett_src, ett_dst, agg_tt, E_TT, stream);
    run_sage128(agg_tt, deg_tt, tx, nullptr, t2_wl, t2_wr, t2_bl,
                tx, N_TAXON, true, stream);

    // ---- output projections (no agg path, no relu) ----
    run_proj64(sotu_x, sl_w, sl_b, z_sotu, N_SOTU, stream);
    run_proj64(tx, tl_w, tl_b, z_taxon, N_TAXON, stream);

    // ---- edge decoder ----
    decoder_kernel<<<(E_LBL + 31) / 32, 256, 0, stream>>>(
        z_sotu, z_taxon, elbl_row, elbl_col, d1_w, d1_b, d2_w, d2_b,
        (float*)d_out, E_LBL);
}